// PowerSpectrum_70385924047570
// MI455X (gfx1250) — compile-verified
//
#include <hip/hip_runtime.h>
#include <hip/hip_bf16.h>
#include <math.h>

// MI455X / gfx1250, wave32. DFT-as-GEMM power spectrum with radial binning.
//
//   F = W * X * W   (W = 256x256 symmetric DFT matrix, ortho: 1/16 per side)
//   power = Fre^2 + Fim^2 ; bins analytic from log2-spaced edges.

#define NN    256
#define NBINS 20
#define KP    260            // padded K pitch (floats): 260 % 64 = 4 -> no bank conflicts,
                             // 260*4B = 1040B = 16B-aligned rows for float4 traffic
#define TKP   (16 * KP)      // one [16][KP] strip, in floats
#define WAVES 2
#define BLOCK (WAVES * 32)

typedef __attribute__((ext_vector_type(2))) float v2f;
typedef __attribute__((ext_vector_type(4))) float v4f;
typedef __attribute__((ext_vector_type(8))) float v8f;

__device__ __forceinline__ v8f wmma_f32(v2f a, v2f b, v8f c) {
  // D = A(16x4) * B(4x16) + C(16x16), full f32
  return __builtin_amdgcn_wmma_f32_16x16x4_f32(false, a, false, b, (short)0, c,
                                               false, false);
}

// ---------------------------------------------------------------------------
// Init: build DFT matrix (re/im) in workspace, zero the output accumulators.
// Deterministic, runs every launch (graph-capture safe).
// ---------------------------------------------------------------------------
__global__ void ps_init(float* __restrict__ wre, float* __restrict__ wim,
                        float* __restrict__ out) {
  int idx = blockIdx.x * blockDim.x + threadIdx.x;
  if (idx < NN * NN) {
    int j = idx >> 8;
    int k = idx & 255;
    int p = (j * k) & 255;  // exploit periodicity for precision
    float th = -6.283185307179586f * (float)p * (1.0f / 256.0f);
    float s, c;
    __sincosf(th, &s, &c);
    wre[idx] = c * 0.0625f;  // 1/sqrt(256) per 1-D transform
    wim[idx] = s * 0.0625f;
  }
  if (idx < 128 * NBINS) out[idx] = 0.0f;
}

// ---------------------------------------------------------------------------
// Main: one workgroup = (image b, 16-column slice n0). 2 waves, 64 threads.
//   stage 1: T_slice = X[b] @ W[:, n0:n0+16]          (real x complex)
//   stage 2: F_slice = W @ T_slice                     (complex x complex)
//   then power + radial binning, LDS bins -> global atomics.
// ---------------------------------------------------------------------------
__global__ __launch_bounds__(BLOCK) void ps_main(
    const float* __restrict__ X, const float* __restrict__ sums,
    const float* __restrict__ Wre, const float* __restrict__ Wim,
    float* __restrict__ out) {
  // LDS (union-allocated): sT (2 strips) + pool (4 strips) + bins ~= 100 KB
  __shared__ __align__(16) float smem[6 * TKP + NBINS];
  float* sT   = smem;            // [c][n][m] : T slice, transposed so stage-2
                                 //             B-frags read contiguous k
  float* pool = smem + 2 * TKP;  // stage1: Wslice[2 strips] + Xstrip per wave
                                 // stage2: (Wre,Wim) strip per wave
  float* bins = smem + 6 * TKP;

  const int tid  = threadIdx.x;
  const int lane = tid & 31;
  const int wv   = tid >> 5;
  const int half = lane >> 4;   // which 16-lane half of the wave
  const int ml   = lane & 15;   // row (A) / col (B,C) within fragment

  const int b  = blockIdx.x >> 4;         // image 0..127
  const int n0 = (blockIdx.x & 15) << 4;  // column-slice base

  if (tid < NBINS) bins[tid] = 0.0f;

  // ---- stage the W column slice (B of stage 1), transposed: sBW[c][n][k]
  float* sBW = pool;  // 2 strips
  for (int i = tid; i < 2 * NN * 4; i += BLOCK) {
    int c  = i >> 10;
    int r  = i & 1023;
    int k  = r >> 2;
    int j4 = (r & 3) << 2;
    const float* src = (c ? Wim : Wre) + k * NN + n0 + j4;
    v4f v = *(const v4f*)src;
    sBW[c * TKP + (j4 + 0) * KP + k] = v.x;
    sBW[c * TKP + (j4 + 1) * KP + k] = v.y;
    sBW[c * TKP + (j4 + 2) * KP + k] = v.z;
    sBW[c * TKP + (j4 + 3) * KP + k] = v.w;
  }
  __syncthreads();

  // =========================== stage 1 ===================================
  const float* Xb = X + (size_t)b * NN * NN;
  {
    float* Xs = pool + 2 * TKP + wv * TKP;  // per-wave X strip [16][KP]
    for (int mt = wv; mt < 16; mt += WAVES) {
      const int m0 = mt << 4;
      // cooperative (per-wave) strip load, float4-coalesced
      for (int i = lane; i < 16 * (NN / 4); i += 32) {
        int m = i >> 6;
        int q = (i & 63) << 2;
        *(v4f*)&Xs[m * KP + q] = *(const v4f*)(Xb + (m0 + m) * NN + q);
      }
      // same-wave LDS store->load is kept in order by hardware/compiler waits
      v8f cre = {0, 0, 0, 0, 0, 0, 0, 0};
      v8f cim = {0, 0, 0, 0, 0, 0, 0, 0};
#pragma unroll 4
      for (int k0 = 0; k0 < NN; k0 += 4) {
        v2f a   = *(const v2f*)&Xs[ml * KP + k0 + 2 * half];
        v2f bre = *(const v2f*)&sBW[0 * TKP + ml * KP + k0 + 2 * half];
        v2f bim = *(const v2f*)&sBW[1 * TKP + ml * KP + k0 + 2 * half];
        cre = wmma_f32(a, bre, cre);
        cim = wmma_f32(a, bim, cim);
      }
      // store C tile into sT[c][n][m] (row m contiguous -> wide ds stores)
      {
        int mb = m0 + 8 * half;
        v4f lo, hi;
        lo.x = cre[0]; lo.y = cre[1]; lo.z = cre[2]; lo.w = cre[3];
        hi.x = cre[4]; hi.y = cre[5]; hi.z = cre[6]; hi.w = cre[7];
        *(v4f*)&sT[0 * TKP + ml * KP + mb]     = lo;
        *(v4f*)&sT[0 * TKP + ml * KP + mb + 4] = hi;
        lo.x = cim[0]; lo.y = cim[1]; lo.z = cim[2]; lo.w = cim[3];
        hi.x = cim[4]; hi.y = cim[5]; hi.z = cim[6]; hi.w = cim[7];
        *(v4f*)&sT[1 * TKP + ml * KP + mb]     = lo;
        *(v4f*)&sT[1 * TKP + ml * KP + mb + 4] = hi;
      }
    }
  }
  __syncthreads();  // sT complete; pool free for stage-2 reuse

  // =========================== stage 2 ===================================
  {
    float* As = pool + wv * 2 * TKP;  // per-wave [c][16][KP] W row strips
    for (int mt = wv; mt < 16; mt += WAVES) {
      const int m0 = mt << 4;
      for (int i = lane; i < 2 * 16 * (NN / 4); i += 32) {
        int c = i >> 10;
        int r = i & 1023;
        int m = r >> 6;
        int q = (r & 63) << 2;
        const float* src = (c ? Wim : Wre) + (m0 + m) * NN + q;
        *(v4f*)&As[c * TKP + m * KP + q] = *(const v4f*)src;
      }
      v8f crr = {0, 0, 0, 0, 0, 0, 0, 0};  // Wre*Tre
      v8f cii = {0, 0, 0, 0, 0, 0, 0, 0};  // Wim*Tim   (Fre = crr - cii)
      v8f cim = {0, 0, 0, 0, 0, 0, 0, 0};  // Wre*Tim + Wim*Tre
#pragma unroll 2
      for (int k0 = 0; k0 < NN; k0 += 4) {
        v2f are = *(const v2f*)&As[0 * TKP + ml * KP + k0 + 2 * half];
        v2f aim = *(const v2f*)&As[1 * TKP + ml * KP + k0 + 2 * half];
        v2f bre = *(const v2f*)&sT[0 * TKP + ml * KP + k0 + 2 * half];
        v2f bim = *(const v2f*)&sT[1 * TKP + ml * KP + k0 + 2 * half];
        crr = wmma_f32(are, bre, crr);
        cii = wmma_f32(aim, bim, cii);
        cim = wmma_f32(are, bim, cim);
        cim = wmma_f32(aim, bre, cim);
      }
      // power + analytic radial binning (edges: 2^(7*i/20), i=0..20)
#pragma unroll
      for (int v = 0; v < 8; ++v) {
        float fre = crr[v] - cii[v];
        float fim = cim[v];
        float p = fre * fre + fim * fim;
        int h = m0 + v + 8 * half;  // output row  (freq axis 0)
        int w = n0 + ml;            // output col  (freq axis 1)
        float fx = (float)(h < 128 ? h : h - 256);
        float fy = (float)(w < 128 ? w : w - 256);
        float kr2 = fx * fx + fy * fy;
        if (kr2 > 1.0f) {  // k > edges[0] = 1 (strict)
          float t = 0.5f * __log2f(kr2);                 // log2(k)
          int bin = (int)ceilf(t * (20.0f / 7.0f)) - 1;  // (edge_i, edge_i+1]
          if (bin >= 0 && bin < NBINS) atomicAdd(&bins[bin], p);
        }
      }
    }
  }
  __syncthreads();

  if (tid < NBINS) {
    atomicAdd(out + b * NBINS + tid, bins[tid] / sums[tid]);
  }
}

// ---------------------------------------------------------------------------
extern "C" void kernel_launch(void* const* d_in, const int* in_sizes, int n_in,
                              void* d_out, int out_size, void* d_ws,
                              size_t ws_size, hipStream_t stream) {
  const float* x    = (const float*)d_in[0];
  // d_in[1] = bin_masks: unused (bin index computed analytically)
  const float* sums = (const float*)d_in[2];
  float* out = (float*)d_out;

  float* wre = (float*)d_ws;        // 256*256 floats
  float* wim = wre + NN * NN;       // 256*256 floats  (512 KB total)

  ps_init<<<(NN * NN + 255) / 256, 256, 0, stream>>>(wre, wim, out);
  ps_main<<<128 * 16, BLOCK, 0, stream>>>(x, sums, wre, wim, out);
}